// scatter_infer_6889127543370
// MI455X (gfx1250) — compile-verified
//
#include <hip/hip_runtime.h>
#include <cfloat>

#define NSEG 65536
#define NCOL 128

// ---------------------------------------------------------------------------
// CDNA5 async global->LDS path (gfx1250): guarded so either toolchain compiles.
// Builtin signature (from hipcc diagnostic): (AS1 int*, AS3 int*, Imm, Imm).
// ---------------------------------------------------------------------------
#if defined(__gfx1250__) && __has_builtin(__builtin_amdgcn_global_load_async_to_lds_b32) && \
    __has_builtin(__builtin_amdgcn_s_wait_asynccnt)
#define USE_ASYNC_LDS 1
typedef __attribute__((address_space(1))) int gint_t;  // global (device) int*
typedef __attribute__((address_space(3))) int lint_t;  // LDS int*
#else
#define USE_ASYNC_LDS 0
#endif

// ---------------------------------------------------------------------------
// Pass 0: zero counts + cursors (2*NSEG contiguous words)
// ---------------------------------------------------------------------------
__global__ void zero_kernel(unsigned* __restrict__ p, int n) {
  int i = blockIdx.x * blockDim.x + threadIdx.x;
  if (i < n) p[i] = 0u;
}

// ---------------------------------------------------------------------------
// Pass 1: histogram of segment sizes
// ---------------------------------------------------------------------------
__global__ void count_kernel(const int* __restrict__ idx, unsigned* __restrict__ counts, int n) {
  int i = blockIdx.x * blockDim.x + threadIdx.x;
  if (i < n) atomicAdd(&counts[idx[i]], 1u);
}

// ---------------------------------------------------------------------------
// Pass 2: exclusive prefix sum over NSEG counters, single 1024-thread block.
// Each thread owns 64 consecutive counters; block-scan of the 1024 partials.
// ---------------------------------------------------------------------------
__global__ __launch_bounds__(1024) void scan_kernel(const unsigned* __restrict__ counts,
                                                    unsigned* __restrict__ offsets) {
  __shared__ unsigned part[1024];
  const int t = threadIdx.x;
  const int per = NSEG / 1024;  // 64
  const int base = t * per;

  unsigned sum = 0u;
  for (int i = 0; i < per; ++i) sum += counts[base + i];
  part[t] = sum;
  __syncthreads();

  // Hillis-Steele inclusive scan over 1024 partials
  for (int off = 1; off < 1024; off <<= 1) {
    unsigned v = (t >= off) ? part[t - off] : 0u;
    __syncthreads();
    part[t] += v;
    __syncthreads();
  }

  unsigned run = (t == 0) ? 0u : part[t - 1];
  for (int i = 0; i < per; ++i) {
    offsets[base + i] = run;
    run += counts[base + i];
  }
  if (t == 1023) offsets[NSEG] = run;  // total
}

// ---------------------------------------------------------------------------
// Pass 3: scatter row ids into CSR order
// ---------------------------------------------------------------------------
__global__ void fill_kernel(const int* __restrict__ idx, const unsigned* __restrict__ offsets,
                            unsigned* __restrict__ cursor, unsigned* __restrict__ row_ids, int n) {
  int i = blockIdx.x * blockDim.x + threadIdx.x;
  if (i >= n) return;
  int s = idx[i];
  unsigned pos = offsets[s] + atomicAdd(&cursor[s], 1u);
  row_ids[pos] = (unsigned)i;
}

// ---------------------------------------------------------------------------
// Pass 4: per-segment reduction. One block per segment, thread = column.
// Async global->LDS staging, 4 rows in flight per wave, 8 LDS row buffers.
// ---------------------------------------------------------------------------
__global__ __launch_bounds__(NCOL) void seg_reduce_kernel(const float* __restrict__ feat,
                                                          const unsigned* __restrict__ offsets,
                                                          const unsigned* __restrict__ row_ids,
                                                          const int* __restrict__ mode_p,
                                                          float* __restrict__ out) {
  const unsigned s   = blockIdx.x;
  const unsigned col = threadIdx.x;
  const unsigned beg = offsets[s];
  const unsigned end = offsets[s + 1];
  const unsigned n   = end - beg;
  const int mode     = mode_p[0];

  float amax = -FLT_MAX;
  float asum = 0.0f;

#if USE_ASYNC_LDS
  // Each wave owns its 32-column slice of each buffer -> no cross-wave sync.
  __shared__ float buf[8][NCOL];

  unsigned staged = (n < 4u) ? n : 4u;
  for (unsigned j = 0; j < staged; ++j) {
    unsigned r = row_ids[beg + j];
    __builtin_amdgcn_global_load_async_to_lds_b32(
        (gint_t*)(feat + (size_t)r * NCOL + col),
        (lint_t*)&buf[j & 7u][col], 0, 0);
  }
  unsigned k = 0;
  for (; staged < n; ++k) {
    __builtin_amdgcn_s_wait_asynccnt(3);  // oldest of 4 in-flight rows is in LDS
    float v = buf[k & 7u][col];
    amax = fmaxf(amax, v);
    asum += v;
    unsigned r = row_ids[beg + staged];
    __builtin_amdgcn_global_load_async_to_lds_b32(
        (gint_t*)(feat + (size_t)r * NCOL + col),
        (lint_t*)&buf[staged & 7u][col], 0, 0);
    ++staged;
  }
  __builtin_amdgcn_s_wait_asynccnt(0);
  for (; k < n; ++k) {
    float v = buf[k & 7u][col];
    amax = fmaxf(amax, v);
    asum += v;
  }
#else
  // Direct-load fallback: 4 loads in flight via manual unroll.
  unsigned k = beg;
  for (; k + 4u <= end; k += 4u) {
    unsigned r0 = row_ids[k + 0], r1 = row_ids[k + 1];
    unsigned r2 = row_ids[k + 2], r3 = row_ids[k + 3];
    float v0 = feat[(size_t)r0 * NCOL + col];
    float v1 = feat[(size_t)r1 * NCOL + col];
    float v2 = feat[(size_t)r2 * NCOL + col];
    float v3 = feat[(size_t)r3 * NCOL + col];
    amax = fmaxf(fmaxf(amax, v0), fmaxf(v1, fmaxf(v2, v3)));
    asum += (v0 + v1) + (v2 + v3);
  }
  for (; k < end; ++k) {
    unsigned r = row_ids[k];
    float v = feat[(size_t)r * NCOL + col];
    amax = fmaxf(amax, v);
    asum += v;
  }
#endif

  float res;
  if (mode == 1)      res = (n > 0u) ? amax : 0.0f;
  else if (mode == 2) res = asum / fmaxf((float)n, 1.0f);
  else                res = asum;
  out[(size_t)s * NCOL + col] = res;
}

// ---------------------------------------------------------------------------
// Fallback (workspace too small): sentinel + float atomic max via int min/max.
// ---------------------------------------------------------------------------
__global__ void fb_init_kernel(float* __restrict__ out, int total, const int* __restrict__ mode_p) {
  int i = blockIdx.x * blockDim.x + threadIdx.x;
  if (i < total) out[i] = (mode_p[0] == 1) ? -FLT_MAX : 0.0f;
}

__global__ void fb_scatter_kernel(const float* __restrict__ feat, const int* __restrict__ idx,
                                  const int* __restrict__ mode_p, float* __restrict__ out,
                                  long long total) {
  long long e = (long long)blockIdx.x * blockDim.x + threadIdx.x;
  if (e >= total) return;
  int row = (int)(e >> 7);
  int col = (int)(e & (NCOL - 1));
  float v = feat[e];
  float* dst = &out[(size_t)idx[row] * NCOL + col];
  if (mode_p[0] == 1) {
    if (v >= 0.0f) atomicMax((int*)dst, __float_as_int(v));
    else           atomicMin((unsigned*)dst, __float_as_uint(v));
  } else {
    atomicAdd(dst, v);
  }
}

__global__ void fb_fix_kernel(float* __restrict__ out, int total, const int* __restrict__ mode_p) {
  int i = blockIdx.x * blockDim.x + threadIdx.x;
  if (i < total && mode_p[0] == 1 && out[i] == -FLT_MAX) out[i] = 0.0f;
}

// ---------------------------------------------------------------------------
extern "C" void kernel_launch(void* const* d_in, const int* in_sizes, int n_in,
                              void* d_out, int out_size, void* d_ws, size_t ws_size,
                              hipStream_t stream) {
  const float* feat   = (const float*)d_in[0];
  const int*   unqinv = (const int*)d_in[1];
  const int*   mode   = (const int*)d_in[2];
  float*       out    = (float*)d_out;
  const int    N      = in_sizes[1];

  const size_t need = (size_t)(3 * NSEG + 1 + N) * sizeof(unsigned);

  if (ws_size >= need) {
    unsigned* counts  = (unsigned*)d_ws;         // NSEG
    unsigned* cursor  = counts + NSEG;           // NSEG (contiguous with counts)
    unsigned* offsets = cursor + NSEG;           // NSEG + 1
    unsigned* row_ids = offsets + NSEG + 1;      // N

    const int zn = 2 * NSEG;
    zero_kernel<<<(zn + 255) / 256, 256, 0, stream>>>(counts, zn);
    count_kernel<<<(N + 255) / 256, 256, 0, stream>>>(unqinv, counts, N);
    scan_kernel<<<1, 1024, 0, stream>>>(counts, offsets);
    fill_kernel<<<(N + 255) / 256, 256, 0, stream>>>(unqinv, offsets, cursor, row_ids, N);
    seg_reduce_kernel<<<NSEG, NCOL, 0, stream>>>(feat, offsets, row_ids, mode, out);
  } else {
    long long total = (long long)N * NCOL;
    fb_init_kernel<<<(out_size + 255) / 256, 256, 0, stream>>>(out, out_size, mode);
    fb_scatter_kernel<<<(int)((total + 255) / 256), 256, 0, stream>>>(feat, unqinv, mode, out, total);
    fb_fix_kernel<<<(out_size + 255) / 256, 256, 0, stream>>>(out, out_size, mode);
  }
}